// RCNNCell_66778151518251
// MI455X (gfx1250) — compile-verified
//
#include <hip/hip_runtime.h>

typedef float v2f __attribute__((ext_vector_type(2)));
typedef float v8f __attribute__((ext_vector_type(8)));

#define NGRID 1024
#define LDSW  36          // 32 region + 2 pad each side (stride 36: A-side bank-conflict-free)
#define DT_F  0.5f

__device__ __forceinline__ v8f wmma4(v2f a, v2f b, v8f c) {
  // D = A(16x4) * B(4x16) + C, full fp32
  return __builtin_amdgcn_wmma_f32_16x16x4_f32(false, a, false, b, (short)0, c,
                                               false, false);
}

__device__ __forceinline__ float sel5(float a0, float a1, float a2, float a3,
                                      float a4, int j) {
  float r = 0.0f;
  r = (j == 0) ? a0 : r;
  r = (j == 1) ? a1 : r;
  r = (j == 2) ? a2 : r;
  r = (j == 3) ? a3 : r;
  r = (j == 4) ? a4 : r;
  return r;
}

// Row-derivative (d1, axis-2), col-derivative (d2, axis-3) and Laplacian of a
// 16x16 tile at local origin (r0,c0) of a padded LDS field, as banded matmuls
// (5 chunks of K=4 over a 20-wide window).
__device__ __forceinline__ void stencil3(const float* F, int r0, int c0,
                                         const v2f* Cd, const v2f* Cl,
                                         int l16, int half,
                                         v8f& d1, v8f& d2, v8f& lp) {
  v8f z = {0.f, 0.f, 0.f, 0.f, 0.f, 0.f, 0.f, 0.f};
  d1 = z; d2 = z; lp = z;
#pragma unroll
  for (int i = 0; i < 5; ++i) {
    const int kk = 4 * i;
    // A-side field slab: lane holds F[r0+l16][c0-2+kk+2*half + {0,1}] (pad=+2)
    v2f a = *(const v2f*)&F[(r0 + l16 + 2) * LDSW + (c0 + kk + 2 * half)];
    // B-side field slab: lane holds F[r0-2+kk+2*half + {0,1}][c0+l16]
    const int br = r0 + kk + 2 * half;  // (-2 offset cancels +2 pad)
    v2f bv;
    bv.x = F[br * LDSW + (c0 + l16 + 2)];
    bv.y = F[(br + 1) * LDSW + (c0 + l16 + 2)];
    d2 = wmma4(a, Cd[i], d2);   // field on A, band on B  -> col derivative
    d1 = wmma4(Cd[i], bv, d1);  // band on A, field on B  -> row derivative
    lp = wmma4(a, Cl[i], lp);   // Laplacian = lap_x + lap_y, one accumulator
    lp = wmma4(Cl[i], bv, lp);
  }
}

__global__ __launch_bounds__(128) void rk4_burgers(
    const float* __restrict__ h, const float* __restrict__ pRe,
    const float* __restrict__ pUA, const float* __restrict__ pUB,
    const float* __restrict__ pVA, const float* __restrict__ pVB,
    float* __restrict__ out) {
  __shared__ float U0[LDSW * LDSW], V0[LDSW * LDSW];
  __shared__ float Ua[LDSW * LDSW], Va[LDSW * LDSW];
  __shared__ float Ub[LDSW * LDSW], Vb[LDSW * LDSW];

  const int tid  = threadIdx.x;
  const int wid  = tid >> 5;        // wave id 0..3 -> owns WMMA tile `wid`
  const int lane = tid & 31;
  const int l16  = lane & 15;
  const int half = lane >> 4;
  const int tx = blockIdx.x, ty = blockIdx.y, b = blockIdx.z;

  // Zero only the 2-wide pad rings (136 float2 spans per array): the garbage
  // ring never leaks into the kept center, but pads must be benign.
  {
    const v2f zz = {0.f, 0.f};
    for (int i = tid; i < 136; i += 128) {
      int idx;
      if (i < 36)       idx = i;                    // rows 0-1
      else if (i < 72)  idx = 612 + (i - 36);       // rows 34-35
      else if (i < 104) idx = (i - 72 + 2) * 18;    // cols 0-1, rows 2..33
      else              idx = (i - 104 + 2) * 18 + 17;  // cols 34-35
      ((v2f*)U0)[idx] = zz; ((v2f*)V0)[idx] = zz;
      ((v2f*)Ua)[idx] = zz; ((v2f*)Va)[idx] = zz;
      ((v2f*)Ub)[idx] = zz; ((v2f*)Vb)[idx] = zz;
    }
  }

  // Cooperative load of the 32x32 halo region (periodic wrap); the 4x halo
  // redundancy across blocks is absorbed by the 192MB L2 (input fits).
  const size_t cs = (size_t)NGRID * NGRID;
  const float* hu = h + (size_t)b * 2 * cs;
  const float* hv = hu + cs;
  const int gx = (tx * 16 - 8 + lane + NGRID) & (NGRID - 1);
#pragma unroll
  for (int rr8 = 0; rr8 < 8; ++rr8) {
    const int rr = wid * 8 + rr8;
    const int gy = (ty * 16 - 8 + rr + NGRID) & (NGRID - 1);
    U0[(rr + 2) * LDSW + lane + 2] = hu[(size_t)gy * NGRID + gx];
    V0[(rr + 2) * LDSW + lane + 2] = hv[(size_t)gy * NGRID + gx];
  }

  const float Re  = pRe[0];
  const float nu  = 0.001f / Re;  // MU_UP / Re
  const float cUA = pUA[0], cUB = pUB[0], cVA = pVA[0], cVB = pVB[0];

  // Banded coefficient operands; identical per-lane value serves both the
  // A side (row deriv) and B side (col deriv): tap(kk + 2*half + e - l16).
  const float ivdx = 100.0f, ivdx2 = 10000.0f;
  v2f Cd[5], Cl[5];
#pragma unroll
  for (int i = 0; i < 5; ++i) {
    const int j0 = 4 * i + 2 * half - l16;
    Cd[i].x = sel5(1.f/12.f, -8.f/12.f, 0.f, 8.f/12.f, -1.f/12.f, j0) * ivdx;
    Cd[i].y = sel5(1.f/12.f, -8.f/12.f, 0.f, 8.f/12.f, -1.f/12.f, j0 + 1) * ivdx;
    Cl[i].x = sel5(-1.f/12.f, 4.f/3.f, -2.5f, 4.f/3.f, -1.f/12.f, j0) * ivdx2;
    Cl[i].y = sel5(-1.f/12.f, 4.f/3.f, -2.5f, 4.f/3.f, -1.f/12.f, j0 + 1) * ivdx2;
  }

  const int r0 = (wid >> 1) * 16;   // this wave's tile origin
  const int c0 = (wid & 1) * 16;
  // C/D layout: element r -> (row r0 + r + 8*half, col c0 + l16)
  const int base = (r0 + 8 * half + 2) * LDSW + (c0 + l16 + 2);

  v8f accU = {0.f,0.f,0.f,0.f,0.f,0.f,0.f,0.f};
  v8f accV = accU;

  __syncthreads();

  // RK4: stage inputs ping-pong U0 -> Ua -> Ub -> Ua; valid region shrinks by
  // 2 per stage while the needed region shrinks by 2 -> no leakage.
  for (int s = 0; s < 4; ++s) {
    const float* US = (s == 0) ? U0 : (s == 2) ? Ub : Ua;
    const float* VS = (s == 0) ? V0 : (s == 2) ? Vb : Va;
    float* UN = (s == 1) ? Ub : Ua;
    float* VN = (s == 1) ? Vb : Va;
    const float w = (s == 1 || s == 2) ? 2.0f : 1.0f;
    const float cstep = (s == 2) ? DT_F : 0.5f * DT_F;
    const bool wr = (s < 3);

    // Point values at this tile's C/D positions, shared by both channels.
    float us8[8], vs8[8];
#pragma unroll
    for (int r = 0; r < 8; ++r) {
      us8[r] = US[base + r * LDSW];
      vs8[r] = VS[base + r * LDSW];
    }

#pragma unroll
    for (int ch = 0; ch < 2; ++ch) {
      const float* FS = ch ? VS : US;
      const float* F0 = ch ? V0 : U0;
      float* FN       = ch ? VN : UN;
      const float ca  = ch ? cVA : cUA;  // multiplies u * d/dy(F)
      const float cb  = ch ? cVB : cUB;  // multiplies v * d/dx(F)
      v8f d1, d2, lp;
      stencil3(FS, r0, c0, Cd, Cl, l16, half, d1, d2, lp);
#pragma unroll
      for (int r = 0; r < 8; ++r) {
        const int li = base + r * LDSW;
        const float k = nu * lp[r] + ca * us8[r] * d1[r] + cb * vs8[r] * d2[r];
        if (ch) accV[r] += w * k; else accU[r] += w * k;
        if (wr) FN[li] = F0[li] + cstep * k;
      }
    }
    if (s < 3) __syncthreads();
  }

  // Dump weighted-k accumulators (Ub/Vb are dead in stage 4) so the final
  // 16x16 center (straddling the 2x2 tile grid) can be gathered.
#pragma unroll
  for (int r = 0; r < 8; ++r) {
    Ub[base + r * LDSW] = accU[r];
    Vb[base + r * LDSW] = accV[r];
  }
  __syncthreads();

  // Final combine + coalesced store: 512 outputs over 128 threads.
  float* ou = out + (size_t)b * 2 * cs;
  float* ov = ou + cs;
#pragma unroll
  for (int e4 = 0; e4 < 4; ++e4) {
    const int e   = e4 * 128 + tid;     // 0..511
    const int chn = e >> 8;             // uniform per iteration
    const int idx = e & 255;
    const int row = 8 + (idx >> 4);
    const int col = 8 + (idx & 15);
    const int li = (row + 2) * LDSW + (col + 2);
    const float sum  = chn ? Vb[li] : Ub[li];
    const float base0 = chn ? V0[li] : U0[li];
    float* op = chn ? ov : ou;
    const int gy  = ty * 16 + row - 8;
    const int gxo = tx * 16 + col - 8;
    op[(size_t)gy * NGRID + gxo] = base0 + (DT_F / 6.0f) * sum;
  }
}

extern "C" void kernel_launch(void* const* d_in, const int* in_sizes, int n_in,
                              void* d_out, int out_size, void* d_ws,
                              size_t ws_size, hipStream_t stream) {
  (void)in_sizes; (void)n_in; (void)out_size; (void)d_ws; (void)ws_size;
  const float* h  = (const float*)d_in[0];
  const float* Re = (const float*)d_in[1];
  const float* UA = (const float*)d_in[2];
  const float* UB = (const float*)d_in[3];
  const float* VA = (const float*)d_in[4];
  const float* VB = (const float*)d_in[5];
  float* out = (float*)d_out;
  dim3 grid(NGRID / 16, NGRID / 16, 8);
  rk4_burgers<<<grid, dim3(128), 0, stream>>>(h, Re, UA, UB, VA, VB, out);
}